// channel_Encoding_Block_60516089201189
// MI455X (gfx1250) — compile-verified
//
#include <hip/hip_runtime.h>

typedef float v2f_ __attribute__((ext_vector_type(2)));
typedef float v8f_ __attribute__((ext_vector_type(8)));

// ---------------------------------------------------------------------------
// Wave-level fp32 GEMM: C(32 x 16*NT) = A(32x32) @ B(32 x 16*NT) via
// V_WMMA_F32_16X16X4_F32. All LDS traffic goes through one shared array `sm`
// passed by reference so addrspace(3) is provable -> plain ds_load/ds_store.
// A-operand: lane n(=lane&15) holds row M; group g(=lane>>4) holds K {2g,2g+1}.
// B-operand: lane n holds column N; group g holds rows {2g,2g+1} of K slice.
// C/D: vgpr r, lane (g,n) -> [M = mt*16 + 8g + r][N = nt*16 + n].
// ---------------------------------------------------------------------------
template <int SZ, int NT>
__device__ __forceinline__ void wave_gemm32(float (&sm)[SZ], int aOff, int lda,
                                            int bOff, int ldb,
                                            v8f_ (&acc)[2][NT]) {
  const int lane = threadIdx.x & 31;
  const int g = lane >> 4;
  const int n = lane & 15;
#pragma unroll
  for (int mt = 0; mt < 2; ++mt) {
#pragma unroll
    for (int nt = 0; nt < NT; ++nt) {
      v8f_ c = {0.f, 0.f, 0.f, 0.f, 0.f, 0.f, 0.f, 0.f};
#pragma unroll
      for (int kk = 0; kk < 8; ++kk) {
        const int k0 = kk * 4 + 2 * g;
        v2f_ a, b;
        a[0] = sm[aOff + (mt * 16 + n) * lda + k0];
        a[1] = sm[aOff + (mt * 16 + n) * lda + k0 + 1];
        b[0] = sm[bOff + k0 * ldb + nt * 16 + n];
        b[1] = sm[bOff + (k0 + 1) * ldb + nt * 16 + n];
        c = __builtin_amdgcn_wmma_f32_16x16x4_f32(false, a, false, b,
                                                  (short)0, c, false, false);
      }
      acc[mt][nt] = c;
    }
  }
}

template <int SZ, int NT>
__device__ __forceinline__ void store_tiles32(const v8f_ (&acc)[2][NT],
                                              float (&sm)[SZ], int cOff,
                                              int ldc) {
  const int lane = threadIdx.x & 31;
  const int g = lane >> 4;
  const int n = lane & 15;
#pragma unroll
  for (int mt = 0; mt < 2; ++mt)
#pragma unroll
    for (int nt = 0; nt < NT; ++nt)
#pragma unroll
      for (int r = 0; r < 8; ++r)
        sm[cOff + (mt * 16 + 8 * g + r) * ldc + nt * 16 + n] = acc[mt][nt][r];
}

// ---------------------------------------------------------------------------
// One full MHSA + residual + LayerNorm + sigmoid for a (T=32, D=32) sequence,
// one wave32. sm[seqOff..] holds input embedding on entry, sigmoid(LN(...))
// on exit. Wperm columns pre-permuted to [Q | K | V], feature f = h*4+d.
// Attention: S = Q.K^T and O = P.V both on WMMA, softmax (exp) on VALU.
// scrOff slab (32x33) is S->P scratch; sums[32] holds per-row 1/sum.
// ---------------------------------------------------------------------------
template <int SZ>
__device__ __forceinline__ void mhsa_ln_sig(float (&sm)[SZ], int seqOff,
                                            int qkvOff, int attOff, int scrOff,
                                            int wpOff, int w0Off, int gbOff,
                                            int sumOff) {
  const int lane = threadIdx.x & 31;
  const int g = lane >> 4;
  const int n = lane & 15;
  const int t = lane;

  // qkv = seq @ Wqvk^T   (32x32 @ 32x96, permuted columns -> [Q|K|V])
  {
    v8f_ acc[2][6];
    wave_gemm32<SZ, 6>(sm, seqOff, 33, wpOff, 96, acc);
    store_tiles32<SZ, 6>(acc, sm, qkvOff, 96);
  }
  __syncthreads();

  for (int h = 0; h < 8; ++h) {
    // ---- S_h = Q_h . K_h^T : 4 WMMAs (K=4), operands direct from qkv LDS.
    {
      v8f_ s[2][2];
#pragma unroll
      for (int mt = 0; mt < 2; ++mt)
#pragma unroll
        for (int nt = 0; nt < 2; ++nt) {
          v2f_ a, b;
          a[0] = sm[qkvOff + (mt * 16 + n) * 96 + 4 * h + 2 * g];
          a[1] = sm[qkvOff + (mt * 16 + n) * 96 + 4 * h + 2 * g + 1];
          b[0] = sm[qkvOff + (nt * 16 + n) * 96 + 32 + 4 * h + 2 * g];
          b[1] = sm[qkvOff + (nt * 16 + n) * 96 + 32 + 4 * h + 2 * g + 1];
          v8f_ c0 = {0.f, 0.f, 0.f, 0.f, 0.f, 0.f, 0.f, 0.f};
          s[mt][nt] = __builtin_amdgcn_wmma_f32_16x16x4_f32(
              false, a, false, b, (short)0, c0, false, false);
        }
      store_tiles32<SZ, 2>(s, sm, scrOff, 33);
    }
    __syncthreads();

    // ---- softmax row pass (lane t owns row t); P written back in place,
    //      1/sum -> sums[t]. scale = DH^-0.5 = 0.5 folded here.
    {
      const int row = scrOff + t * 33;
      float a[32];
      float mx = -3.4e38f;
#pragma unroll
      for (int si = 0; si < 32; ++si) {
        const float v = 0.5f * sm[row + si];
        a[si] = v;
        mx = fmaxf(mx, v);
      }
      float sum = 0.f;
#pragma unroll
      for (int si = 0; si < 32; ++si) {
        const float p = __expf(a[si] - mx);
        sum += p;
        sm[row + si] = p;
      }
      sm[sumOff + t] = 1.0f / sum;
    }
    __syncthreads();

    // ---- O_h = P . V_h : 2x8 WMMAs; only N columns 0..3 are meaningful
    //      (B over-reads neighbouring qkv/att columns, results discarded).
    {
      v8f_ o[2];
#pragma unroll
      for (int mt = 0; mt < 2; ++mt) {
        v8f_ c = {0.f, 0.f, 0.f, 0.f, 0.f, 0.f, 0.f, 0.f};
#pragma unroll
        for (int kk = 0; kk < 8; ++kk) {
          const int k0 = kk * 4 + 2 * g;
          v2f_ a, b;
          a[0] = sm[scrOff + (mt * 16 + n) * 33 + k0];
          a[1] = sm[scrOff + (mt * 16 + n) * 33 + k0 + 1];
          b[0] = sm[qkvOff + k0 * 96 + 64 + 4 * h + n];
          b[1] = sm[qkvOff + (k0 + 1) * 96 + 64 + 4 * h + n];
          c = __builtin_amdgcn_wmma_f32_16x16x4_f32(false, a, false, b,
                                                    (short)0, c, false, false);
        }
        o[mt] = c;
      }
#pragma unroll
      for (int mt = 0; mt < 2; ++mt)
#pragma unroll
        for (int r = 0; r < 8; ++r) {
          const int rowM = mt * 16 + 8 * g + r;
          const float inv = sm[sumOff + rowM];  // uniform within lane group
          if (n < 4)
            sm[attOff + rowM * 33 + 4 * h + n] = o[mt][r] * inv;
        }
    }
    __syncthreads();
  }

  // o = att_out @ W0^T
  {
    v8f_ acc[2][2];
    wave_gemm32<SZ, 2>(sm, attOff, 33, w0Off, 32, acc);
    store_tiles32<SZ, 2>(acc, sm, scrOff, 33);
  }
  __syncthreads();

  // resid += emb
#pragma unroll
  for (int q = 0; q < 32; ++q) {
    const int p = q * 32 + lane;
    const int r = p >> 5, j = p & 31;
    sm[scrOff + r * 33 + j] += sm[seqOff + r * 33 + j];
  }
  __syncthreads();

  // LayerNorm (feature axis) + sigmoid; one row per lane, write into seq.
  {
    const int row = scrOff + t * 33;
    float mu = 0.f;
#pragma unroll
    for (int j = 0; j < 32; ++j) mu += sm[row + j];
    mu *= (1.0f / 32.0f);
    float var = 0.f;
#pragma unroll
    for (int j = 0; j < 32; ++j) {
      const float d = sm[row + j] - mu;
      var += d * d;
    }
    var *= (1.0f / 32.0f);
    const float rstd = rsqrtf(var + 1e-5f);
#pragma unroll
    for (int j = 0; j < 32; ++j) {
      const float y = (sm[row + j] - mu) * rstd * sm[gbOff + j] + sm[gbOff + 32 + j];
      sm[seqOff + t * 33 + j] = 1.0f / (1.0f + __expf(-y));
    }
  }
  __syncthreads();
}

// ---------------- Kernel 1 LDS layout (floats) -----------------------------
#define K1_E 0
#define K1_SEQ 1056
#define K1_QKV 2112
#define K1_ATT 5184
#define K1_SCR 6240
#define K1_POST 7296
#define K1_WP 8320
#define K1_W0 11392
#define K1_GB 12416
#define K1_SUM 12480
#define K1_SZ 12512

__global__ __launch_bounds__(32) void ceb_stage1(
    const float* __restrict__ x, const float* __restrict__ pos_y,
    const float* __restrict__ Wqvk_x, const float* __restrict__ W0_x,
    const float* __restrict__ g_x, const float* __restrict__ b_x,
    float* __restrict__ sigY) {
  __shared__ float sm[K1_SZ];

  const int b = blockIdx.x;
  const int lane = threadIdx.x;
  const float* xg = x + (size_t)(b * 3 + 2) * 32768;

#pragma unroll
  for (int q = 0; q < 32; ++q) {
    const int p = q * 32 + lane;
    const int j = p >> 5, d = p & 31;
    sm[K1_POST + d * 32 + j] = pos_y[2 * 1024 + j * 32 + d];  // posT[d][j]
    sm[K1_E + (p >> 5) * 33 + (p & 31)] =
        xg[(p >> 5) * 1024 + (p & 31) * 32 + 31];  // x[b,2,t,d,31]
  }
  sm[K1_GB + lane] = g_x[lane];
  sm[K1_GB + 32 + lane] = b_x[lane];
  __syncthreads();

  {  // emb_last = A @ posT (in place into sE)
    v8f_ acc[2][2];
    wave_gemm32<K1_SZ, 2>(sm, K1_E, 33, K1_POST, 32, acc);
    __syncthreads();
    store_tiles32<K1_SZ, 2>(acc, sm, K1_E, 33);
  }
  __syncthreads();

#pragma unroll 1
  for (int c = 0; c < 3; ++c) {
#pragma unroll
    for (int q = 0; q < 96; ++q) {
      const int p = q * 32 + lane;
      const int j = p / 96, op = p % 96;
      const int sel = op >> 5, e = op & 31, hh = e >> 2, dd = e & 3;
      const int o = dd * 24 + sel * 8 + hh;  // original qkv row index
      sm[K1_WP + p] = Wqvk_x[c * 3072 + o * 32 + j];
    }
#pragma unroll
    for (int q = 0; q < 32; ++q) {
      const int p = q * 32 + lane;
      sm[K1_W0 + p] = W0_x[c * 1024 + (p & 31) * 32 + (p >> 5)];  // W0[o][f]
    }
#pragma unroll
    for (int q = 0; q < 33; ++q)
      sm[K1_SEQ + q * 32 + lane] = sm[K1_E + q * 32 + lane];
    __syncthreads();

    mhsa_ln_sig<K1_SZ>(sm, K1_SEQ, K1_QKV, K1_ATT, K1_SCR, K1_WP, K1_W0,
                       K1_GB, K1_SUM);

#pragma unroll
    for (int q = 0; q < 32; ++q) {
      const int p = q * 32 + lane;
      const int t = p >> 5, j = p & 31;
      sigY[((size_t)(b * 3 + c) * 32 + t) * 32 + j] = sm[K1_SEQ + t * 33 + j];
    }
    __syncthreads();
  }
}

// ---------------- Kernel 2 LDS layout (floats) -----------------------------
#define K2_SLAB 0
#define K2_QKV 4224
#define K2_ATT 7296
#define K2_SCR 8352
#define K2_POST 9408
#define K2_WP 10432
#define K2_W0 13504
#define K2_GB 14528
#define K2_SY 14592
#define K2_SUM 14720
#define K2_SZ 14752

__global__ __launch_bounds__(32) void ceb_stage2(
    const float* __restrict__ x, const float* __restrict__ pos_y,
    const float* __restrict__ Wqvk_y, const float* __restrict__ W0_y,
    const float* __restrict__ g_y, const float* __restrict__ b_y,
    const float* __restrict__ sigY, float* __restrict__ out) {
  __shared__ float sm[K2_SZ];

  const int blk = blockIdx.x;  // 0 .. B*3*8-1
  const int iq = blk & 7;
  const int c = (blk >> 3) % 3;
  const int b = blk / 24;
  const int lane = threadIdx.x;
  const int i0 = iq * 4;

  const float* xg = x + (size_t)(b * 3 + c) * 32768;

#pragma unroll
  for (int q = 0; q < 32; ++q) {
    const int p = q * 32 + lane;
    const int j = p >> 5, d = p & 31;
    sm[K2_POST + d * 32 + j] = pos_y[c * 1024 + j * 32 + d];
    sm[K2_W0 + p] = W0_y[c * 1024 + (p & 31) * 32 + (p >> 5)];
  }
#pragma unroll
  for (int q = 0; q < 96; ++q) {
    const int p = q * 32 + lane;
    const int j = p / 96, op = p % 96;
    const int sel = op >> 5, e = op & 31, hh = e >> 2, dd = e & 3;
    const int o = dd * 24 + sel * 8 + hh;
    sm[K2_WP + p] = Wqvk_y[c * 3072 + o * 32 + j];
  }
  sm[K2_GB + lane] = g_y[lane];
  sm[K2_GB + 32 + lane] = b_y[lane];
#pragma unroll
  for (int q = 0; q < 4; ++q) {
    const int p = q * 32 + lane;
    const int ii = p >> 5, t = p & 31;
    sm[K2_SY + p] = sigY[((size_t)(b * 3 + c) * 32 + (i0 + ii)) * 32 + t];
  }
  // Load the x quad, float4 over the contiguous i axis, scatter to 4 slabs.
#pragma unroll
  for (int q = 0; q < 32; ++q) {
    const int p = q * 32 + lane;
    const int t = p >> 5, d = p & 31;
    const float4 v =
        *reinterpret_cast<const float4*>(xg + t * 1024 + d * 32 + i0);
    sm[K2_SLAB + 0 * 1056 + t * 33 + d] = v.x;
    sm[K2_SLAB + 1 * 1056 + t * 33 + d] = v.y;
    sm[K2_SLAB + 2 * 1056 + t * 33 + d] = v.z;
    sm[K2_SLAB + 3 * 1056 + t * 33 + d] = v.w;
  }
  __syncthreads();

#pragma unroll 1
  for (int ip = 0; ip < 4; ++ip) {
    const int seqOff = K2_SLAB + ip * 1056;
    {  // emb = x_slice @ posT (in place)
      v8f_ acc[2][2];
      wave_gemm32<K2_SZ, 2>(sm, seqOff, 33, K2_POST, 32, acc);
      __syncthreads();
      store_tiles32<K2_SZ, 2>(acc, sm, seqOff, 33);
    }
    __syncthreads();
    mhsa_ln_sig<K2_SZ>(sm, seqOff, K2_QKV, K2_ATT, K2_SCR, K2_WP, K2_W0,
                       K2_GB, K2_SUM);
  }

  // out[b,c,t,j,i] = sigY[b,c,i,t] * sigmoid(LN)[i][t,j] * x[b,c,t,j,i]
  float* og = out + (size_t)(b * 3 + c) * 32768;
#pragma unroll
  for (int q = 0; q < 32; ++q) {
    const int p = q * 32 + lane;
    const int t = p >> 5, j = p & 31;
    const float4 xv =
        *reinterpret_cast<const float4*>(xg + t * 1024 + j * 32 + i0);
    float4 ov;
    ov.x = sm[K2_SY + 0 * 32 + t] * sm[K2_SLAB + 0 * 1056 + t * 33 + j] * xv.x;
    ov.y = sm[K2_SY + 1 * 32 + t] * sm[K2_SLAB + 1 * 1056 + t * 33 + j] * xv.y;
    ov.z = sm[K2_SY + 2 * 32 + t] * sm[K2_SLAB + 2 * 1056 + t * 33 + j] * xv.z;
    ov.w = sm[K2_SY + 3 * 32 + t] * sm[K2_SLAB + 3 * 1056 + t * 33 + j] * xv.w;
    *reinterpret_cast<float4*>(og + t * 1024 + j * 32 + i0) = ov;
  }
}

extern "C" void kernel_launch(void* const* d_in, const int* in_sizes, int n_in,
                              void* d_out, int out_size, void* d_ws,
                              size_t ws_size, hipStream_t stream) {
  const float* x = (const float*)d_in[0];
  const float* pos_y = (const float*)d_in[1];
  const float* Wqvk_y = (const float*)d_in[2];
  const float* W0_y = (const float*)d_in[3];
  const float* g_y = (const float*)d_in[4];
  const float* b_y = (const float*)d_in[5];
  const float* Wqvk_x = (const float*)d_in[6];
  const float* W0_x = (const float*)d_in[7];
  const float* g_x = (const float*)d_in[8];
  const float* b_x = (const float*)d_in[9];
  float* out = (float*)d_out;
  float* sigY = (float*)d_ws;  // B*3*32*32 floats = 3 MB

  const int B = in_sizes[0] / (3 * 32 * 32 * 32);  // 256

  ceb_stage1<<<B, 32, 0, stream>>>(x, pos_y, Wqvk_x, W0_x, g_x, b_x, sigY);
  ceb_stage2<<<B * 3 * 8, 32, 0, stream>>>(x, pos_y, Wqvk_y, W0_y, g_y, b_y,
                                           sigY, out);
}